// ExactContinuousSBM_74646531604678
// MI455X (gfx1250) — compile-verified
//
#include <hip/hip_runtime.h>
#include <math.h>

// ---- problem constants (from reference) ----
#define BN      8192
#define DN      32
#define KN      64
#define NSTEP   255
#define BETA_MIN_C 0.1f
#define BETA_MAX_C 20.0f
#define T_MIN_C 1e-4f
#define T_MAX_C 1.0f
#define TWO_PI_C 6.28318530717958647692f

// block config: 4 waves * 32 lanes, 16 rows per wave -> 64 rows per block
#define THREADS 128
#define ROWS_PER_BLOCK 64
#define XSTR 33   // padded stride for [K][D]-style LDS arrays
#define RSTR 68   // padded stride for per-wave R[16][64] buffer

typedef __attribute__((ext_vector_type(2))) float v2f;
typedef __attribute__((ext_vector_type(8))) float v8f;

__global__ __launch_bounds__(THREADS)
void ExactContinuousSBM_74646531604678_kernel(const float* __restrict__ x_init,
                                              const float* __restrict__ centers,
                                              const float* __restrict__ stds,
                                              const float* __restrict__ weights,
                                              const float* __restrict__ noise,
                                              float* __restrict__ out) {
    __shared__ float iv_lds[KN * XSTR];          // 1/v[k][d]
    __shared__ float bm_lds[KN * XSTR];          // m[k][d]/v[k][d]
    __shared__ float ck_lds[KN];                 // per-k logc constant
    __shared__ float lw_lds[KN];                 // log weights
    __shared__ float x_lds[ROWS_PER_BLOCK * XSTR];
    __shared__ float r_lds[4 * 16 * RSTR];       // per-wave responsibilities R[16][64]

    const int tid  = threadIdx.x;
    const int lane = tid & 31;
    const int wave = tid >> 5;
    const int blockRow0 = blockIdx.x * ROWS_PER_BLOCK;

    const int m16  = lane & 15;                  // M index / N index within tile
    const int kb   = (lane < 16) ? 0 : 2;        // K sub-offset of this half-wave
    const int kcol = lane & 15;
    const int rowHalf = (lane < 16) ? 0 : 8;     // C/D layout row half

    // ---- init: x_init -> LDS, log(weights) -> LDS ----
    for (int i = tid; i < ROWS_PER_BLOCK * DN; i += THREADS) {
        int r = i >> 5, d = i & 31;
        x_lds[r * XSTR + d] = x_init[(size_t)(blockRow0 + r) * DN + d];
    }
    if (tid < KN) lw_lds[tid] = __logf(weights[tid]);
    __syncthreads();

    const float step_h = (T_MAX_C - T_MIN_C) / (float)NSTEP;
    const float dt = -step_h;                    // constant negative dt
    float* rbase = &r_lds[wave * 16 * RSTR];
    const int xrow0 = wave * 16;

    for (int s = 0; s < NSTEP; ++s) {
        const float t   = T_MAX_C - (float)s * step_h;
        const float bt  = BETA_MIN_C + (BETA_MAX_C - BETA_MIN_C) * t;
        const float Bi  = BETA_MIN_C * t + 0.5f * (BETA_MAX_C - BETA_MIN_C) * t * t;
        const float eB  = __expf(-Bi);
        const float seB = sqrtf(eB);
        const float omeB = 1.0f - eB;

        // ---- stage A: per-step mixture tables (threads 0..63, one k each) ----
        if (tid < KN) {
            const int k = tid;
            float qs = 0.0f, ls = 0.0f;
            #pragma unroll 4
            for (int d = 0; d < DN; ++d) {
                float c  = centers[k * DN + d];
                float sd = stds[k * DN + d];
                float v  = eB * sd * sd + omeB;
                float ivv = 1.0f / v;
                float m  = seB * c;
                iv_lds[k * XSTR + d] = ivv;
                bm_lds[k * XSTR + d] = m * ivv;
                qs += m * m * ivv;
                ls += __logf(TWO_PI_C * v);
            }
            ck_lds[k] = lw_lds[k] - 0.5f * qs - 16.0f * ls;  // (D/2)*sum(log) term
        }
        __syncthreads();

        // ---- prefetch next step's noise slice (one 128B line per thread) ----
        if ((s + 1 < NSTEP) && (tid < 64)) {
            const float* pf = noise + ((size_t)(s + 1) * BN + blockRow0) * DN + tid * 32;
            __builtin_prefetch(pf, 0, 3);
        }

        // ---- gather this lane's A-fragment x elements (d mod 4 in {kb, kb+1}) ----
        float xr[16];
        {
            const int xb = (xrow0 + m16) * XSTR;
            #pragma unroll
            for (int kk = 0; kk < 8; ++kk) {
                xr[2 * kk]     = x_lds[xb + 4 * kk + kb];
                xr[2 * kk + 1] = x_lds[xb + 4 * kk + kb + 1];
            }
        }

        // ---- GEMM1: logc[16 rows][64 k] = [-0.5x^2, x] . [iv ; m*iv]^T ----
        v8f acc[4] = {};
        #pragma unroll
        for (int kk = 0; kk < 16; ++kk) {
            v2f a;
            const float* bmat;
            int crow;
            if (kk < 8) {
                float x0 = xr[2 * kk], x1 = xr[2 * kk + 1];
                a[0] = -0.5f * x0 * x0;
                a[1] = -0.5f * x1 * x1;
                bmat = iv_lds;
                crow = 4 * kk + kb;
            } else {
                a[0] = xr[2 * (kk - 8)];
                a[1] = xr[2 * (kk - 8) + 1];
                bmat = bm_lds;
                crow = 4 * (kk - 8) + kb;
            }
            #pragma unroll
            for (int jt = 0; jt < 4; ++jt) {
                const int kg = jt * 16 + kcol;   // output k column
                v2f b;
                b[0] = bmat[kg * XSTR + crow];
                b[1] = bmat[kg * XSTR + crow + 1];
                acc[jt] = __builtin_amdgcn_wmma_f32_16x16x4_f32(
                    false, a, false, b, (short)0, acc[jt], false, false);
            }
        }

        // ---- softmax over k (per row), write R into per-wave LDS buffer ----
        float ck4[4];
        #pragma unroll
        for (int jt = 0; jt < 4; ++jt) ck4[jt] = ck_lds[jt * 16 + kcol];

        #pragma unroll
        for (int g = 0; g < 8; ++g) {
            float l0 = acc[0][g] + ck4[0];
            float l1 = acc[1][g] + ck4[1];
            float l2 = acc[2][g] + ck4[2];
            float l3 = acc[3][g] + ck4[3];
            float mx = fmaxf(fmaxf(l0, l1), fmaxf(l2, l3));
            mx = fmaxf(mx, __shfl_xor(mx, 1, 32));
            mx = fmaxf(mx, __shfl_xor(mx, 2, 32));
            mx = fmaxf(mx, __shfl_xor(mx, 4, 32));
            mx = fmaxf(mx, __shfl_xor(mx, 8, 32));
            float p0 = __expf(l0 - mx), p1 = __expf(l1 - mx);
            float p2 = __expf(l2 - mx), p3 = __expf(l3 - mx);
            float sm = p0 + p1 + p2 + p3;
            sm += __shfl_xor(sm, 1, 32);
            sm += __shfl_xor(sm, 2, 32);
            sm += __shfl_xor(sm, 4, 32);
            sm += __shfl_xor(sm, 8, 32);
            float inv = 1.0f / sm;
            const int row = g + rowHalf;
            rbase[row * RSTR + 0 * 16 + kcol] = p0 * inv;
            rbase[row * RSTR + 1 * 16 + kcol] = p1 * inv;
            rbase[row * RSTR + 2 * 16 + kcol] = p2 * inv;
            rbase[row * RSTR + 3 * 16 + kcol] = p3 * inv;
        }
        asm volatile("" ::: "memory");  // keep R stores before R loads (same wave, DS in-order)

        // ---- GEMM2: S1 = R . iv  (16x32),  S2 = R . (m*iv)  (16x32) ----
        v8f s1a[2] = {}, s2a[2] = {};
        #pragma unroll
        for (int kk = 0; kk < 16; ++kk) {
            v2f a;
            a[0] = rbase[m16 * RSTR + 4 * kk + kb];
            a[1] = rbase[m16 * RSTR + 4 * kk + kb + 1];
            const int krow = 4 * kk + kb;
            #pragma unroll
            for (int dtl = 0; dtl < 2; ++dtl) {
                const int d = dtl * 16 + kcol;
                v2f b1, b2;
                b1[0] = iv_lds[krow * XSTR + d];
                b1[1] = iv_lds[(krow + 1) * XSTR + d];
                b2[0] = bm_lds[krow * XSTR + d];
                b2[1] = bm_lds[(krow + 1) * XSTR + d];
                s1a[dtl] = __builtin_amdgcn_wmma_f32_16x16x4_f32(
                    false, a, false, b1, (short)0, s1a[dtl], false, false);
                s2a[dtl] = __builtin_amdgcn_wmma_f32_16x16x4_f32(
                    false, a, false, b2, (short)0, s2a[dtl], false, false);
            }
        }

        // ---- Euler–Maruyama update (elements in C/D layout, wave-private rows) ----
        const float nscale = sqrtf(bt * (-dt));
        const float* zrow = noise + ((size_t)s * BN + blockRow0) * DN;
        #pragma unroll
        for (int dtl = 0; dtl < 2; ++dtl) {
            #pragma unroll
            for (int g = 0; g < 8; ++g) {
                const int row = g + rowHalf;
                const int col = dtl * 16 + kcol;
                const int xi = (xrow0 + row) * XSTR + col;
                float xo = x_lds[xi];
                float sc = s2a[dtl][g] - xo * s1a[dtl][g];       // score
                float drift = -bt * (0.5f * xo + sc);
                float z = zrow[(size_t)(xrow0 + row) * DN + col];
                x_lds[xi] = xo + drift * dt + nscale * z;
            }
        }
        __syncthreads();  // protect iv/bm/ck for next step's stage A
    }

    // ---- write final x ----
    for (int i = tid; i < ROWS_PER_BLOCK * DN; i += THREADS) {
        int r = i >> 5, d = i & 31;
        out[(size_t)(blockRow0 + r) * DN + d] = x_lds[r * XSTR + d];
    }
}

extern "C" void kernel_launch(void* const* d_in, const int* in_sizes, int n_in,
                              void* d_out, int out_size, void* d_ws, size_t ws_size,
                              hipStream_t stream) {
    (void)in_sizes; (void)n_in; (void)out_size; (void)d_ws; (void)ws_size;
    const float* x_init  = (const float*)d_in[0];
    const float* centers = (const float*)d_in[1];
    const float* stds    = (const float*)d_in[2];
    const float* weights = (const float*)d_in[3];
    const float* noise   = (const float*)d_in[4];
    float* out = (float*)d_out;

    dim3 grid(BN / ROWS_PER_BLOCK);   // 128 blocks
    dim3 block(THREADS);              // 4 waves
    ExactContinuousSBM_74646531604678_kernel<<<grid, block, 0, stream>>>(
        x_init, centers, stds, weights, noise, out);
}